// Sinkhorn_20675972563813
// MI455X (gfx1250) — compile-verified
//
#include <hip/hip_runtime.h>

// Problem constants (from reference): x,y are (NT, NTRACES, NCH) fp32.
#define NT     4096
#define NPAIR  4096              // NTRACES*NCH, pair p = trace*4+ch, elem at t*NPAIR+p
#define SEG    16                // segment length (= one WMMA tile of t)
#define NSEG   (NT / SEG)        // 256 segments per pair
#define SEGBUF (NPAIR * NSEG)    // 1,048,576 floats per per-segment array
#define NP3    (NSEG * (NPAIR / 16))  // 65536 phase-3 wave tasks

typedef __attribute__((ext_vector_type(2))) float v2f;
typedef __attribute__((ext_vector_type(8))) float v8f;

// ---------------------------------------------------------------------------
// Phase 1: per (pair, 16-seg): s = sum(relu), q = sum of local inclusive cumsum
// lanes <-> pairs => every load is a 128B contiguous wave transaction (HBM pass)
// ---------------------------------------------------------------------------
__global__ __launch_bounds__(256) void ph1_segsums(
    const float* __restrict__ x, const float* __restrict__ y,
    float* __restrict__ sx, float* __restrict__ qx,
    float* __restrict__ sy, float* __restrict__ qy)
{
    int w    = (blockIdx.x * 256 + threadIdx.x) >> 5;  // 32768 waves
    int lane = threadIdx.x & 31;
    int pg   = w & 127;          // pair group (32 pairs)
    int s    = w >> 7;           // segment 0..255
    int p    = pg * 32 + lane;
    size_t base = (size_t)s * SEG * NPAIR + p;
    float cx = 0.f, qxa = 0.f, cy = 0.f, qya = 0.f;
#pragma unroll
    for (int j = 0; j < SEG; ++j) {
        float rx = fmaxf(x[base + (size_t)j * NPAIR], 0.0f);
        float ry = fmaxf(y[base + (size_t)j * NPAIR], 0.0f);
        cx += rx; qxa += cx;
        cy += ry; qya += cy;
    }
    int o = s * NPAIR + p;
    sx[o] = cx; qx[o] = qxa; sy[o] = cy; qy[o] = qya;
}

// ---------------------------------------------------------------------------
// Phase 2: per pair: exclusive prefix of segment sums -> o_s, and
// A = sum_t a_t = sum_s (16*o_s + q_s)  -> invA.  Block = 32 pairs, 8 waves,
// each wave owns a 32-segment chunk; chunk sums exchanged through LDS.
// ---------------------------------------------------------------------------
__global__ __launch_bounds__(256) void ph2_scan(
    const float* __restrict__ sx, const float* __restrict__ qx,
    const float* __restrict__ sy, const float* __restrict__ qy,
    float* __restrict__ ox, float* __restrict__ oy,
    float* __restrict__ invA, float* __restrict__ invB)
{
    __shared__ float csx[8][32], csy[8][32], pAx[8][32], pAy[8][32];
    int lane = threadIdx.x & 31;
    int w    = threadIdx.x >> 5;          // chunk id 0..7
    int p    = blockIdx.x * 32 + lane;    // pair

    // step A: chunk sums
    float cx = 0.f, cy = 0.f;
    for (int i = 0; i < 32; ++i) {
        int s = w * 32 + i;
        cx += sx[s * NPAIR + p];
        cy += sy[s * NPAIR + p];
    }
    csx[w][lane] = cx; csy[w][lane] = cy;
    __syncthreads();

    // step B: chunk-exclusive offset, then serial scan within chunk
    float offx = 0.f, offy = 0.f;
    for (int c = 0; c < w; ++c) { offx += csx[c][lane]; offy += csy[c][lane]; }
    float pax = 0.f, pay = 0.f;
    for (int i = 0; i < 32; ++i) {
        int s = w * 32 + i, o = s * NPAIR + p;
        float svx = sx[o], svy = sy[o];
        ox[o] = offx; oy[o] = offy;
        pax += 16.0f * offx + qx[o];
        pay += 16.0f * offy + qy[o];
        offx += svx; offy += svy;
    }
    pAx[w][lane] = pax; pAy[w][lane] = pay;
    __syncthreads();

    if (w == 0) {
        float Ax = 0.f, Ay = 0.f;
        for (int c = 0; c < 8; ++c) { Ax += pAx[c][lane]; Ay += pAy[c][lane]; }
        invA[p] = 1.0f / Ax;
        invB[p] = 1.0f / Ay;
    }
}

// ---------------------------------------------------------------------------
// Phase 3: cumsum of a 16x16 (t x pair) tile as a triangular-matrix multiply:
// D = T * relu(X), T lower-triangular ones, via 4 chained V_WMMA_F32_16X16X4_F32.
// A layout: lane&15 = row M; VGPR0/1 hold K = 2*half + {0,1}.
// B layout (mirrored over N): lane&15 = col N (pair); VGPR0/1 = K = 2*half+{0,1}.
// D layout: M = v + 8*half, N = lane&15.
// ---------------------------------------------------------------------------
__device__ inline v8f tri_cumsum_tile(const float* __restrict__ src,
                                      int t0, int pair, int half, int m)
{
    v8f c = {};
#pragma unroll
    for (int j = 0; j < 4; ++j) {
        int kbase = 4 * j + 2 * half;     // global K of this lane's two K-slots
        v2f a, b;
        a.x = (kbase     <= m) ? 1.0f : 0.0f;   // T[m][kbase]
        a.y = (kbase + 1 <= m) ? 1.0f : 0.0f;   // T[m][kbase+1]
        size_t addr = (size_t)(t0 + kbase) * NPAIR + pair;
        b.x = fmaxf(src[addr], 0.0f);
        b.y = fmaxf(src[addr + NPAIR], 0.0f);
        c = __builtin_amdgcn_wmma_f32_16x16x4_f32(
                /*neg_a=*/false, a, /*neg_b=*/false, b,
                /*c_mod=*/(short)0, c, /*reuse_a=*/false, /*reuse_b=*/false);
    }
    return c;
}

__global__ __launch_bounds__(256) void ph3_loss(
    const float* __restrict__ x, const float* __restrict__ y,
    const float* __restrict__ ox, const float* __restrict__ oy,
    const float* __restrict__ invA, const float* __restrict__ invB,
    float* __restrict__ partials)
{
    int w    = (blockIdx.x * 256 + threadIdx.x) >> 5;  // 65536 wave tasks
    int lane = threadIdx.x & 31;
    int pbi  = w & ((NPAIR / 16) - 1);   // 16-pair block
    int s    = w >> 8;                   // segment (16 t's)
    int half = lane >> 4;
    int n    = lane & 15;
    int pair = pbi * 16 + n;
    int t0   = s * SEG;

    v8f Dx = tri_cumsum_tile(x, t0, pair, half, n);
    v8f Dy = tri_cumsum_tile(y, t0, pair, half, n);

    float oxv = ox[s * NPAIR + pair];
    float oyv = oy[s * NPAIR + pair];
    float iA  = invA[pair];
    float iB  = invB[pair];

    float part = 0.f;
#pragma unroll
    for (int v = 0; v < 8; ++v) {
        float av = (Dx[v] + oxv) * iA;      // a_t / A
        float bv = (Dy[v] + oyv) * iB;      // b_t / B
        float d  = av - bv;
        float tg = (float)(t0 + v + 8 * half);
        part += tg * d * d;
    }
    // wave32 fixed-order reduction
#pragma unroll
    for (int off = 16; off > 0; off >>= 1)
        part += __shfl_xor(part, off, 32);
    if (lane == 0) partials[w] = part;
}

// ---------------------------------------------------------------------------
// Phase 4: deterministic fixed-order double reduction of 65536 partials
// ---------------------------------------------------------------------------
__global__ __launch_bounds__(256) void ph4_reduce(
    const float* __restrict__ partials, float* __restrict__ out, int nelem)
{
    __shared__ double sm[256];
    double acc = 0.0;
    for (int i = threadIdx.x; i < nelem; i += 256) acc += (double)partials[i];
    sm[threadIdx.x] = acc;
    __syncthreads();
    for (int off = 128; off > 0; off >>= 1) {
        if ((int)threadIdx.x < off) sm[threadIdx.x] += sm[threadIdx.x + off];
        __syncthreads();
    }
    if (threadIdx.x == 0) out[0] = (float)sm[0];
}

// ---------------------------------------------------------------------------
extern "C" void kernel_launch(void* const* d_in, const int* in_sizes, int n_in,
                              void* d_out, int out_size, void* d_ws, size_t ws_size,
                              hipStream_t stream)
{
    (void)in_sizes; (void)n_in; (void)out_size; (void)ws_size;
    const float* x = (const float*)d_in[0];
    const float* y = (const float*)d_in[1];
    float* outp = (float*)d_out;

    float* wsf = (float*)d_ws;
    float* sx = wsf + 0 * (size_t)SEGBUF;
    float* qx = wsf + 1 * (size_t)SEGBUF;
    float* sy = wsf + 2 * (size_t)SEGBUF;
    float* qy = wsf + 3 * (size_t)SEGBUF;
    float* ox = wsf + 4 * (size_t)SEGBUF;
    float* oy = wsf + 5 * (size_t)SEGBUF;
    float* invA = wsf + 6 * (size_t)SEGBUF;
    float* invB = invA + NPAIR;
    float* partials = invB + NPAIR;

    // Phase 1: 32768 waves (HBM streaming pass, populates 192MB L2)
    ph1_segsums<<<4096, 256, 0, stream>>>(x, y, sx, qx, sy, qy);
    // Phase 2: per-pair segment scan + 1/A
    ph2_scan<<<128, 256, 0, stream>>>(sx, qx, sy, qy, ox, oy, invA, invB);
    // Phase 3: 65536 waves, WMMA triangular-scan tiles (L2-resident pass)
    ph3_loss<<<8192, 256, 0, stream>>>(x, y, ox, oy, invA, invB, partials);
    // Phase 4: deterministic reduction to scalar
    ph4_reduce<<<1, 256, 0, stream>>>(partials, outp, NP3);
}